// SelfAttentionPooling_36747740184625
// MI455X (gfx1250) — compile-verified
//
#include <hip/hip_runtime.h>
#include <hip/hip_bf16.h>
#include <stdint.h>

// ---------------------------------------------------------------------------
// SelfAttentionPooling for MI455X (gfx1250).
//   scores = sigmoid(x @ w + b);  out[g,:] += scores[n] * x[n,:]  (batch sorted)
// HBM floor: 512 MB / 23.3 TB/s ~= 22 us -> memory bound, fp32 end to end.
// Persistent blocks (256 thr = 8 wave32), double-buffered TDM pipeline:
//   - Each block strides over 64-node tiles. While tile i is processed, wave 0
//     has already issued tensor_load_to_lds for tile i+1 into the other LDS
//     buffer, so HBM stays saturated during compute regardless of how many
//     blocks the scheduler co-resides per WGP.
//   - TDM hardware row padding (+8 dwords per 256) -> 264-float LDS rows,
//     bank-conflict-free for both phases.
//   - Phase 1: gate scores via wave32 dot + shfl_xor reduction + sigmoid,
//     packed (score,lg) meta = 8 B/node -> one ds_load_b128 per quad later.
//   - Phase 2: segment-sum as masked GEMM on the matrix cores:
//     D[16 graph-slots x 16 dims] += S[16x4] * X[4x16], v_wmma_f32_16x16x4_f32.
//     Sliding 16-id window, scalar readlane-driven retry loop handles arbitrary
//     id jumps (empty graphs) exactly; atomicAdd flush on window advance.
// ---------------------------------------------------------------------------

typedef __attribute__((ext_vector_type(2))) float v2f;
typedef __attribute__((ext_vector_type(8))) float v8f;
typedef __attribute__((ext_vector_type(4))) unsigned int u32x4;
typedef __attribute__((ext_vector_type(8))) unsigned int u32x8;

#define DIMS 256
#define ROWP 264          // padded LDS row stride in floats (256 + 8)
#define NT   64           // nodes per tile
#define THREADS 256
#define MAXBLOCKS 2048

__global__ __launch_bounds__(THREADS)
void sap_wmma_kernel(const float* __restrict__ x,
                     const int*   __restrict__ batch,
                     const float* __restrict__ attn_w,
                     const float* __restrict__ attn_b,
                     float* __restrict__ out,
                     int N, int G, int nTiles)
{
    extern __shared__ float smem[];
    float* xsb0 = smem;                       // tile buffer 0: NT*ROWP floats
    float* xsb1 = smem + NT * ROWP;           // tile buffer 1
    float* mt0  = smem + 2 * NT * ROWP;       // meta buffer 0: NT x {score,lg}
    float* mt1  = mt0 + 2 * NT;               // meta buffer 1

    const int tid  = threadIdx.x;
    const int lane = tid & 31;
    const int w    = tid >> 5;

    // Issue one TDM tile load (wave 0 only; block-uniform operands).
    auto issue_tdm = [&](int tile, float* xsDst) {
        const int nbT  = tile * NT;
        const int remT = N - nbT;
        const int NTvT = remT < NT ? remT : NT;
        const uint64_t ga   = (uint64_t)(uintptr_t)(x + (size_t)nbT * DIMS);
        const uint32_t lds0 = (uint32_t)(uintptr_t)xsDst; // low 32b = LDS offset
        u32x4 g0;
        g0.x = 1u;                                        // count=1 user descriptor
        g0.y = lds0;                                      // lds_addr
        g0.z = (uint32_t)ga;                              // global_addr[31:0]
        g0.w = (uint32_t)((ga >> 32) & 0x01FFFFFFu) | (2u << 30); // addr[56:32]|type=2
        u32x8 g1;
        // data_size=4B(2), pad_enable, pad_interval=7 (256 dw), pad_amount=7 (8 dw)
        g1.s0 = (2u << 16) | (1u << 20) | (7u << 22) | (7u << 25);
        g1.s1 = ((uint32_t)DIMS & 0xFFFFu) << 16;         // tensor_dim0 lo16 = 256
        g1.s2 = ((uint32_t)remT & 0xFFFFu) << 16;         // dim0 hi16=0 | dim1 lo16
        g1.s3 = (((uint32_t)remT >> 16) & 0xFFFFu) | ((uint32_t)DIMS << 16);
        g1.s4 = (uint32_t)NTvT;                           // tile_dim1 | tile_dim2=0
        g1.s5 = (uint32_t)DIMS;                           // tensor_dim0_stride lo32
        g1.s6 = 0u;
        g1.s7 = 0u;
        const u32x4 gz = {0u, 0u, 0u, 0u};                // groups 2/3 unused (2D)
        asm volatile("tensor_load_to_lds %0, %1, %2, %3"
                     :: "s"(g0), "s"(g1), "s"(gz), "s"(gz) : "memory");
    };

    // Per-block constants (amortized over all tiles of this block).
    const float4 wa = *(const float4*)(attn_w + 4 * lane);
    const float4 wb = *(const float4*)(attn_w + 128 + 4 * lane);
    const float  bias = attn_b[0];
    const int  row = lane & 15;       // A row (graph slot) / B column index
    const bool hi  = lane >= 16;      // hi lanes: K=2,3 of A / rows 2,3 of B
    const int  Dw  = w * 32;          // wave w owns dims [32w, 32w+32)
    const int  col = lane & 15;
    const int  naOff = hi ? 2 : 0;
    const v8f VZ = {0.f,0.f,0.f,0.f,0.f,0.f,0.f,0.f};

    const int tile0 = blockIdx.x;
    const int tstep = gridDim.x;
    if (tile0 >= nTiles) return;

    if (w == 0) issue_tdm(tile0, xsb0);       // prime the pipeline

    int parity = 0;
    for (int tile = tile0; tile < nTiles; tile += tstep, parity ^= 1) {
        float* xs    = parity ? xsb1 : xsb0;
        float* meta  = parity ? mt1  : mt0;
        int*   metai = (int*)meta;
        const int nb  = tile * NT;
        const int rem = N - nb;
        const int NTv = rem < NT ? rem : NT;
        const int gfirst = batch[nb];

        if (w == 0) __builtin_amdgcn_s_wait_tensorcnt(0);  // tile's data arrived
        __syncthreads();   // also: all waves done with the *other* buffer

        // overlap: DMA of the next tile streams during this tile's compute
        const int ntile = tile + tstep;
        if (w == 0 && ntile < nTiles) issue_tdm(ntile, parity ? xsb0 : xsb1);

        if (tid < NTv) metai[2 * tid + 1] = batch[nb + tid] - gfirst;

        // ---- Phase 1: gate scores, NT/8 = 8 nodes per wave ----
        {
            const int nEnd = (w * 8 + 8) < NTv ? (w * 8 + 8) : NTv;
            for (int n = w * 8; n < nEnd; ++n) {
                const float* xr = xs + n * ROWP;
                const float4 xa = *(const float4*)(xr + 4 * lane);
                const float4 xb = *(const float4*)(xr + 128 + 4 * lane);
                float p = xa.x*wa.x + xa.y*wa.y + xa.z*wa.z + xa.w*wa.w
                        + xb.x*wb.x + xb.y*wb.y + xb.z*wb.z + xb.w*wb.w;
                #pragma unroll
                for (int off = 16; off > 0; off >>= 1) p += __shfl_xor(p, off, 32);
                if (lane == 0) meta[2 * n] = 1.0f / (1.0f + __expf(-(p + bias)));
            }
        }
        __syncthreads();

        // ---- Phase 2: masked-WMMA segment-sum over this tile ----
        v8f acc0 = VZ, acc1 = VZ;
        int slot_base = 0;            // uniform (SGPR) sliding window base

        auto flush = [&](int sbase) {
            const int Mb = hi ? 8 : 0;   // C/D: VGPR r -> M=r (lanes<16) / r+8
            #pragma unroll
            for (int r = 0; r < 8; ++r) {
                const int gid = gfirst + sbase + Mb + r;
                if (gid < G) {
                    const float v0 = acc0[r], v1 = acc1[r];
                    if (v0 != 0.0f) atomicAdd(out + (size_t)gid * DIMS + Dw + col, v0);
                    if (v1 != 0.0f) atomicAdd(out + (size_t)gid * DIMS + Dw + 16 + col, v1);
                }
            }
            acc0 = VZ; acc1 = VZ;
        };

        // Masked WMMA for current window; scalar retry loop advances the window
        // across arbitrary id gaps (window passes are disjoint -> exact).
        auto quad = [&](float sa, int lgA, float sb, int lgB, v2f B0, v2f B1) {
            const int mx = __builtin_amdgcn_readlane(lgB, 31);  // quad max lg
            for (;;) {
                v2f A;
                A.x = (lgA - slot_base == row) ? sa : 0.0f;
                A.y = (lgB - slot_base == row) ? sb : 0.0f;
                acc0 = __builtin_amdgcn_wmma_f32_16x16x4_f32(false, A, false, B0,
                                                             (short)0, acc0, false, false);
                acc1 = __builtin_amdgcn_wmma_f32_16x16x4_f32(false, A, false, B1,
                                                             (short)0, acc1, false, false);
                if (mx < slot_base + 16) break;                 // common case
                flush(slot_base);
                const int c0 = __builtin_amdgcn_readlane(lgA, 0);
                const int c1 = __builtin_amdgcn_readlane(lgB, 0);
                const int c2 = __builtin_amdgcn_readlane(lgA, 16);
                const int bound = slot_base + 16;               // first uncovered id
                slot_base = (c0 >= bound) ? c0 : (c1 >= bound) ? c1
                          : (c2 >= bound) ? c2 : mx;
            }
        };

        const int tMain = NTv & ~3;
        {   // main loop: full quads, pointer-increment addressing, no clamps
            const float* mp = meta + 2 * naOff;               // 16B-aligned
            const float* bp = xs + naOff * ROWP + Dw + col;
            for (int t = 0; t < tMain; t += 4) {
                const float4 m = *(const float4*)mp;  mp += 8; // {sa,lgA,sb,lgB}
                const float* b0 = bp;
                const float* b1 = bp + ROWP;          bp += 4 * ROWP;
                v2f B0, B1;
                B0.x = b0[0];  B0.y = b1[0];
                B1.x = b0[16]; B1.y = b1[16];
                quad(m.x, __float_as_int(m.y), m.z, __float_as_int(m.w), B0, B1);
            }
        }
        if (tMain < NTv) {  // ragged tail: clamped indices, invalid scores zeroed
            const int t   = tMain;
            const int na  = t + naOff, nb1 = na + 1;
            const int nac = na  < NTv ? na  : NTv - 1;
            const int nbc = nb1 < NTv ? nb1 : NTv - 1;
            const float sa = (na  < NTv) ? meta[2 * nac] : 0.0f;
            const float sb = (nb1 < NTv) ? meta[2 * nbc] : 0.0f;
            const int  lgA = metai[2 * nac + 1];
            const int  lgB = metai[2 * nbc + 1];  // lane31 -> lg[NTv-1] = valid max
            const float* r0 = xs + nac * ROWP + Dw + col;
            const float* r1 = xs + nbc * ROWP + Dw + col;
            v2f B0, B1;
            B0.x = r0[0];  B0.y = r1[0];
            B1.x = r0[16]; B1.y = r1[16];
            quad(sa, lgA, sb, lgB, B0, B1);
        }
        flush(slot_base);
    }
}

extern "C" void kernel_launch(void* const* d_in, const int* in_sizes, int n_in,
                              void* d_out, int out_size, void* d_ws, size_t ws_size,
                              hipStream_t stream) {
    const float* x      = (const float*)d_in[0];
    const int*   batch  = (const int*)  d_in[1];
    const float* attn_w = (const float*)d_in[2];
    const float* attn_b = (const float*)d_in[3];
    float* out = (float*)d_out;

    const int N = in_sizes[0] / DIMS;       // 500000
    const int G = out_size / DIMS;          // 4096

    hipMemsetAsync(d_out, 0, (size_t)out_size * sizeof(float), stream);

    const int nTiles = (N + NT - 1) / NT;
    const int blocks = nTiles < MAXBLOCKS ? nTiles : MAXBLOCKS;
    const size_t shmem = (size_t)2 * NT * ROWP * sizeof(float)  // 2 padded tiles
                       + (size_t)2 * NT * 8;                    // 2 meta buffers
    hipFuncSetAttribute(reinterpret_cast<const void*>(&sap_wmma_kernel),
                        hipFuncAttributeMaxDynamicSharedMemorySize, (int)shmem);
    sap_wmma_kernel<<<blocks, THREADS, shmem, stream>>>(x, batch, attn_w, attn_b,
                                                        out, N, G, nTiles);
}